// BezierRenderer_37082747634600
// MI455X (gfx1250) — compile-verified
//
#include <hip/hip_runtime.h>
#include <hip/hip_bf16.h>
#include <stdint.h>

// ---------------------------------------------------------------------------
// CDNA5 / gfx1250 Bezier soft-disk renderer
//   Stage A: per-path center + avg radius via V_WMMA_F32_16X16X4_F32
//   Stage B: per-pixel sequential alpha compositing, path table staged in LDS
//            via GLOBAL_LOAD_ASYNC_TO_LDS_B128 (ASYNCcnt) when available.
// ---------------------------------------------------------------------------

#define CANVAS 224
#define NPIX   (CANVAS * CANVAS)     // 50176 = 196 * 256 exactly
#define NPATHS 1024
#define INV_SOFT 50.0f               // 1 / 0.02
#define LOG2E 1.442695040888963f

typedef __attribute__((ext_vector_type(2))) float v2f;
typedef __attribute__((ext_vector_type(8))) float v8f;
typedef __attribute__((ext_vector_type(4))) int   v4i;

#define AS1 __attribute__((address_space(1)))
#define AS3 __attribute__((address_space(3)))

#if defined(__has_builtin)
#if __has_builtin(__builtin_amdgcn_global_load_async_to_lds_b128)
#define HAVE_ASYNC_LDS 1
#endif
#endif

// ---------------------------------------------------------------------------
// Kernel A: one wave handles 2 paths.
//   B (4x16 f32):  B[k][n] = cp[path(n)][seg(n)][k][coord(n)]
//       n = seg*2+coord within a path, cols 0-7 -> path p0, 8-15 -> p0+1
//   A (16x4 f32):  rows = 16 t-samples of the cubic Bezier basis (tile T)
//   D (16x16 f32): sample points; col parity = x/y, paired via shfl_xor(1)
// ---------------------------------------------------------------------------
__global__ __launch_bounds__(128) void bezier_path_stats(
    const float* __restrict__ pcp,      // (P,4,4,2)
    const float* __restrict__ colors,   // (P,4)
    float* __restrict__ table,          // (P,8): cx,cy,rad,alpha, r,g,b,0
    int n_paths)
{
    const int lane = threadIdx.x & 31;
    const int wave = threadIdx.x >> 5;
    const int p0 = (blockIdx.x * 4 + wave) * 2;
    if (p0 >= n_paths) return;          // wave-uniform exit: EXEC stays all-1s

    const int n     = lane & 15;        // column 0..15
    const int hlf   = lane >> 4;        // half: K/M offset selector
    const int path  = p0 + (n >> 3);
    const int seg   = (n >> 1) & 3;
    const int coord = n & 1;
    const int kb    = hlf * 2;          // this lane's K base (0 or 2)

    // Load this lane's path control points: cf[s*2+0]={x0,y0,x1,y1}(k=0,1),
    // cf[s*2+1]={x2,y2,x3,y3}(k=2,3)
    float4 cf[8];
    {
        const float4* src = (const float4*)(pcp + (size_t)path * 32);
#pragma unroll
        for (int i = 0; i < 8; ++i) cf[i] = src[i];
    }

    // Column sums of the basis over t_i = i/49, i=0..48 (mean commutes).
    float S0 = 0.f, S1 = 0.f, S2 = 0.f, S3 = 0.f;
#pragma unroll 1
    for (int i = 0; i < 49; ++i) {
        float t = (float)i * (1.0f / 49.0f);
        float mt = 1.0f - t;
        S0 += mt * mt * mt;
        S1 += 3.0f * mt * mt * t;
        S2 += 3.0f * mt * t * t;
        S3 += t * t * t;
    }

    // Center of this lane's path (mean of 196 sample points).
    float cx = 0.f, cy = 0.f;
#pragma unroll
    for (int sg = 0; sg < 4; ++sg) {
        float4 q0 = cf[sg * 2 + 0], q1 = cf[sg * 2 + 1];
        cx += S0 * q0.x + S1 * q0.z + S2 * q1.x + S3 * q1.z;
        cy += S0 * q0.y + S1 * q0.w + S2 * q1.y + S3 * q1.w;
    }
    cx *= (1.0f / 196.0f);
    cy *= (1.0f / 196.0f);

    // B operand: lane holds K = kb, kb+1 for its column.
    v2f B;
    {
        float4 bq = cf[seg * 2 + (kb >> 1)];
        B.x = coord ? bq.y : bq.x;      // k = kb   (even)
        B.y = coord ? bq.w : bq.z;      // k = kb+1 (odd)
    }

    // Radius: 4 WMMA t-tiles; each D row is one sample component.
    float rsum = 0.f;
#pragma unroll
    for (int T = 0; T < 4; ++T) {
        int trow = T * 16 + (lane & 15);         // A-row this lane supplies
        float tv = (float)trow * (1.0f / 49.0f);
        float mt = 1.0f - tv;
        float bas0 = mt * mt * mt;
        float bas1 = 3.0f * mt * mt * tv;
        float bas2 = 3.0f * mt * tv * tv;
        float bas3 = tv * tv * tv;
        bool rv = (trow < 49);
        v2f A;
        A.x = rv ? (kb ? bas2 : bas0) : 0.0f;    // K = kb
        A.y = rv ? (kb ? bas3 : bas1) : 0.0f;    // K = kb+1
        v8f C = {};
        v8f D = __builtin_amdgcn_wmma_f32_16x16x4_f32(
            false, A, false, B, (short)0, C, false, false);
#pragma unroll
        for (int r = 0; r < 8; ++r) {
            float mine = D[r];
            float other = __shfl_xor(mine, 1, 32); // pair x<->y columns
            float xv = coord ? other : mine;
            float yv = coord ? mine : other;
            int t = T * 16 + r + hlf * 8;          // sample index of this row
            if (t < 49) {
                float dx = xv - cx, dy = yv - cy;
                rsum += __builtin_amdgcn_sqrtf(dx * dx + dy * dy);
            }
        }
    }

    // Reduce over this path's lane group {n<8 | n>=8} x both halves.
    // Each x/y lane pair holds identical sums -> divide by 2.
    float s = rsum;
    s += __shfl_xor(s, 16, 32);
    s += __shfl_xor(s, 1, 32);
    s += __shfl_xor(s, 2, 32);
    s += __shfl_xor(s, 4, 32);
    float radius = s * (1.0f / (2.0f * 196.0f));

    if (lane == 0 || lane == 8) {                 // one writer per path
        const float* c = colors + (size_t)path * 4;
        float4 t0 = make_float4(cx, cy, radius, c[3]);
        float4 t1 = make_float4(c[0], c[1], c[2], 0.0f);
        float4* dst = (float4*)(table + (size_t)path * 8);
        dst[0] = t0;
        dst[1] = t1;
    }
}

// ---------------------------------------------------------------------------
// Kernel B: one thread per pixel, sequential composite over all paths.
// Path table (32 KB) staged once into LDS (async DMA when available);
// per-iteration reads are same-address LDS broadcasts.
// ---------------------------------------------------------------------------
__global__ __launch_bounds__(256) void bezier_composite(
    const float* __restrict__ table,    // (NPATHS, 8)
    float* __restrict__ out,            // (3, 224, 224)
    int n_paths)
{
    __shared__ float4 lds4[NPATHS * 2]; // 32 KB of 320 KB/WGP

    const int tid = threadIdx.x;
    const int nvec = n_paths * 2;       // float4 entries to stage

#if defined(HAVE_ASYNC_LDS)
#pragma unroll
    for (int i = 0; i < (NPATHS * 2) / 256; ++i) {
        int idx = i * 256 + tid;
        if (idx < nvec) {
            __builtin_amdgcn_global_load_async_to_lds_b128(
                (AS1 v4i*)(table + (size_t)idx * 4),
                (AS3 v4i*)(&lds4[idx]),
                0, 0);
        }
    }
#if __has_builtin(__builtin_amdgcn_s_wait_asynccnt)
    __builtin_amdgcn_s_wait_asynccnt(0);
#else
    asm volatile("s_wait_asynccnt 0x0" ::: "memory");
#endif
#else
#pragma unroll
    for (int i = 0; i < (NPATHS * 2) / 256; ++i) {
        int idx = i * 256 + tid;
        if (idx < nvec) lds4[idx] = ((const float4*)table)[idx];
    }
#endif
    __syncthreads();

    const int pix = blockIdx.x * 256 + tid;
    const int x = pix % CANVAS;
    const int y = pix / CANVAS;
    const float px = (float)x * (1.0f / (CANVAS - 1));
    const float py = (float)y * (1.0f / (CANVAS - 1));

    float cr = 1.0f, cg = 1.0f, cb = 1.0f;

    const int np = (n_paths < NPATHS) ? n_paths : NPATHS;
#pragma unroll 4
    for (int p = 0; p < np; ++p) {
        float4 t0 = lds4[p * 2 + 0];     // cx, cy, rad, alpha (LDS broadcast)
        float4 t1 = lds4[p * 2 + 1];     // r, g, b, 0
        float dx = px - t0.x;
        float dy = py - t0.y;
        float d = __builtin_amdgcn_sqrtf(dx * dx + dy * dy);
        // m = alpha * sigmoid((rad - d)/0.02) = alpha / (1 + 2^((d-rad)*50*log2e))
        float z = (d - t0.z) * (INV_SOFT * LOG2E);
        float m = t0.w * __builtin_amdgcn_rcpf(1.0f + __builtin_amdgcn_exp2f(z));
        cr = fmaf(m, t1.x - cr, cr);     // canvas*(1-m) + c*m
        cg = fmaf(m, t1.y - cg, cg);
        cb = fmaf(m, t1.z - cb, cb);
    }

    if (pix < NPIX) {
        out[0 * NPIX + pix] = cr;
        out[1 * NPIX + pix] = cg;
        out[2 * NPIX + pix] = cb;
    }
}

// ---------------------------------------------------------------------------
extern "C" void kernel_launch(void* const* d_in, const int* in_sizes, int n_in,
                              void* d_out, int out_size, void* d_ws, size_t ws_size,
                              hipStream_t stream)
{
    const float* pcp = (const float*)d_in[0];     // (P,4,4,2) f32
    const float* colors = (const float*)d_in[1];  // (P,4) f32
    float* out = (float*)d_out;                   // (3,224,224) f32
    float* table = (float*)d_ws;                  // P*8 floats (32 KB for P=1024)

    const int n_paths = in_sizes[0] / 32;         // 4*4*2 floats per path
    const int waves = (n_paths + 1) / 2;          // 2 paths per wave32
    const int blocksA = (waves + 3) / 4;          // 4 waves (128 thr) per block

    hipLaunchKernelGGL(bezier_path_stats, dim3(blocksA), dim3(128), 0, stream,
                       pcp, colors, table, n_paths);

    const int blocksB = (NPIX + 255) / 256;       // 196 blocks
    hipLaunchKernelGGL(bezier_composite, dim3(blocksB), dim3(256), 0, stream,
                       table, out, n_paths);
}